// ForwardForwardCoutingAutoencoder_90443421319692
// MI455X (gfx1250) — compile-verified
//
#include <hip/hip_runtime.h>
#include <stdint.h>

// ---------------------------------------------------------------------------
// ForwardForwardCountingLayer, CDNA5 (gfx1250).
//  - wave32: one output neuron per wave; 32 lanes split in_f (coalesced loads),
//    cross-lane min/max via __shfl_xor (5 steps).
//  - x row staged to LDS with global_load_async_to_lds_b128 (ASYNCcnt) +
//    s_wait_asynccnt, shared by all 8 waves in the block.
//  - Edge thresholds thr[o,i] = c1/(c0+c1)*2^32 precomputed once per launch
//    (batch-independent) into L2-resident scratch.
//  - Per-edge uniforms from 4 parallel LCG streams per lane (2 VALU/edge),
//    seeded deterministically per (b, o, lane, slot, layer) via mix32.
//  - T-norm vs T-conorm: wave-uniform branch into templated min/max scans;
//    min trees written as chains so LLVM can fold them into v_min3_num_f32.
// ---------------------------------------------------------------------------

#define FFC_WAVES 8                         // waves per block (256 threads)
#define FFC_NOUT  8                         // outputs per wave
#define FFC_OPB   (FFC_WAVES * FFC_NOUT)    // outputs per block = 64

#define FFC_LCG_A 1664525u
#define FFC_LCG_C 1013904223u

__device__ __forceinline__ uint32_t ffc_mix32(uint32_t x) {
  x ^= x >> 16; x *= 0x7feb352du;
  x ^= x >> 15; x *= 0x846ca68bu;
  x ^= x >> 16;
  return x;
}

// Scan all in_f edges of output o for this wave; returns wave-reduced acc.
template <bool IS_NORM, bool USE_THR>
__device__ __forceinline__ float ffc_scan(
    const float* __restrict__ s_x,
    const float* __restrict__ counts_row,   // counts + o*in_f*2 (if !USE_THR)
    const uint32_t* __restrict__ thr_row,   // thr + o*in_f      (if USE_THR)
    int in_f, int lane, uint32_t seed_base, uint32_t salt)
{
  const float NEUTRAL = IS_NORM ? 10.0f : -10.0f;
  float acc = NEUTRAL;

  // 4 independent LCG streams (one per vector slot) -> 2 VALU per edge,
  // ILP-friendly (no serial dependency across slots).
  uint32_t r0 = ffc_mix32((seed_base + 0u) ^ salt);
  uint32_t r1 = ffc_mix32((seed_base + 1u) ^ salt);
  uint32_t r2 = ffc_mix32((seed_base + 2u) ^ salt);
  uint32_t r3 = ffc_mix32((seed_base + 3u) ^ salt);

  if constexpr (USE_THR) {
    // 4 edges per lane per iteration: uint4 thresholds + float4 x values.
    const uint4*  trow = (const uint4*)thr_row;
    const float4* sx4  = (const float4*)s_x;
    const int iters = in_f >> 7;            // in_f / (32 lanes * 4 edges)
#pragma unroll 8
    for (int k = 0; k < iters; ++k) {
      const int vi = (k << 5) + lane;       // float4 / uint4 index
      const uint4  t  = trow[vi];
      const float4 xv = sx4[vi];
      r0 = r0 * FFC_LCG_A + FFC_LCG_C;
      r1 = r1 * FFC_LCG_A + FFC_LCG_C;
      r2 = r2 * FFC_LCG_A + FFC_LCG_C;
      r3 = r3 * FFC_LCG_A + FFC_LCG_C;
      const float s0 = (r0 < t.x) ? xv.x : NEUTRAL;
      const float s1 = (r1 < t.y) ? xv.y : NEUTRAL;
      const float s2 = (r2 < t.z) ? xv.z : NEUTRAL;
      const float s3 = (r3 < t.w) ? xv.w : NEUTRAL;
      if constexpr (IS_NORM) {
        // Left-leaning chain -> two v_min3_num_f32 after ISel folding.
        acc = fminf(fminf(fminf(fminf(acc, s0), s1), s2), s3);
      } else {
        acc = fmaxf(fmaxf(fmaxf(fmaxf(acc, s0), s1), s2), s3);
      }
    }
  } else {
    // Fallback: derive threshold from counts inline (2 edges per float4 load).
    const float4* crow = (const float4*)counts_row;
    const float2* sx2  = (const float2*)s_x;
    const int iters = in_f >> 6;            // in_f / (32 lanes * 2 edges)
#pragma unroll 8
    for (int k = 0; k < iters; ++k) {
      const int vi = (k << 5) + lane;
      const float4 c  = crow[vi];           // (c0,c1) edge0, (c0,c1) edge1
      const float2 xv = sx2[vi];
      r0 = r0 * FFC_LCG_A + FFC_LCG_C;
      r1 = r1 * FFC_LCG_A + FFC_LCG_C;
      // h/2^32 < c1/(c0+c1)  <=>  (float)h*(c0+c1) < c1*2^32   (no divide)
      const bool t0 = ((float)r0) * (c.x + c.y) < c.y * 4294967296.0f;
      const bool t1 = ((float)r1) * (c.z + c.w) < c.w * 4294967296.0f;
      const float s0 = t0 ? xv.x : NEUTRAL;
      const float s1 = t1 ? xv.y : NEUTRAL;
      if constexpr (IS_NORM) acc = fminf(fminf(acc, s0), s1);
      else                   acc = fmaxf(fmaxf(acc, s0), s1);
    }
  }

  // Cross-lane reduction (wave32).
  for (int off = 16; off > 0; off >>= 1) {
    const float other = __shfl_xor(acc, off, 32);
    acc = IS_NORM ? fminf(acc, other) : fmaxf(acc, other);
  }
  return acc;
}

template <bool USE_THR>
__global__ __launch_bounds__(256) void ffc_layer_kernel(
    const float*    __restrict__ x,       // [B, in_f]
    const float*    __restrict__ counts,  // [out_f, in_f, 2]
    const uint32_t* __restrict__ thr,     // [out_f, in_f] (USE_THR only)
    const int*      __restrict__ op,      // [out_f]
    float*          __restrict__ out,     // [B, out_f]
    int in_f, int out_f, uint32_t salt)
{
  __shared__ __align__(16) float s_x[1024];   // max in_f -> 4 KB
  __shared__ float s_out[FFC_OPB];

  const int tid  = threadIdx.x;
  const int lane = tid & 31;
  const int wave = tid >> 5;
  const int b    = blockIdx.x;

  // ---- Stage x[b, :] into LDS with async global->LDS DMA (ASYNCcnt) ------
  const int n16 = in_f >> 2;                  // float4 chunks (<= 256)
  if (tid < n16) {                            // wave-uniform (in_f % 128 == 0)
    uint32_t lds_addr = (uint32_t)(uintptr_t)(&s_x[tid * 4]); // LDS offset
    uint32_t voff     = (uint32_t)tid * 16u;                  // byte offset
    const float* base = x + (size_t)b * (size_t)in_f;
    asm volatile("global_load_async_to_lds_b128 %0, %1, %2"
                 :
                 : "v"(lds_addr), "v"(voff), "s"(base)
                 : "memory");
  }
  // No-op for waves that issued nothing (ASYNCcnt == 0).
  asm volatile("s_wait_asynccnt 0x0" ::: "memory");
  __syncthreads();

  for (int j = 0; j < FFC_NOUT; ++j) {
    const int o = blockIdx.y * FFC_OPB + wave * FFC_NOUT + j;

    // Prefetch next output's table row into cache (global_prefetch_b8).
    if (o + 1 < out_f) {
      if constexpr (USE_THR)
        __builtin_prefetch((const void*)(thr + (size_t)(o + 1) * in_f + lane * 4), 0, 0);
      else
        __builtin_prefetch((const void*)(counts + (size_t)(o + 1) * in_f * 2 + lane * 4), 0, 0);
    }

    const bool is_norm = (op[o] != 0);        // wave-uniform
    // Unique PRNG seed base per (b, o, lane, slot): 4 slots per lane.
    const uint32_t seed_base =
        (((uint32_t)b * (uint32_t)out_f + (uint32_t)o) * 32u + (uint32_t)lane) * 4u;
    const float*    crow = counts + (size_t)o * (size_t)in_f * 2;
    const uint32_t* trow = USE_THR ? (thr + (size_t)o * (size_t)in_f) : nullptr;

    float acc;
    if (is_norm)
      acc = ffc_scan<true,  USE_THR>(s_x, crow, trow, in_f, lane, seed_base, salt);
    else
      acc = ffc_scan<false, USE_THR>(s_x, crow, trow, in_f, lane, seed_base, salt);

    if (lane == 0) {
      const float neutral = is_norm ? 10.0f : -10.0f;
      float res;
      if (acc == neutral) {
        // All sampled edge types were no_edge: force one random edge.
        const uint32_t r = ffc_mix32(((uint32_t)b * (uint32_t)out_f + (uint32_t)o)
                                     ^ (salt + 0x01234567u)) & (uint32_t)(in_f - 1);
        res = s_x[r];
      } else {
        res = acc;
      }
      s_out[wave * FFC_NOUT + j] = res;
    }
  }

  __syncthreads();
  // Coalesced output burst: first 64 threads store the block's outputs.
  if (tid < FFC_OPB) {
    out[(size_t)b * (size_t)out_f + (size_t)blockIdx.y * FFC_OPB + tid] = s_out[tid];
  }
}

// thr[o,i] = c1/(c0+c1) * 2^32 via raw v_rcp_f32 (threshold needs few bits).
__global__ __launch_bounds__(256) void ffc_thr_kernel(
    const float2* __restrict__ counts, uint32_t* __restrict__ thr, int n)
{
  const int i = blockIdx.x * blockDim.x + threadIdx.x;
  if (i < n) {
    const float2 c = counts[i];
    const float p = c.y * __builtin_amdgcn_rcpf(c.x + c.y);
    thr[i] = (uint32_t)fminf(fmaxf(p, 0.0f) * 4294967296.0f, 4294967040.0f);
  }
}

extern "C" void kernel_launch(void* const* d_in, const int* in_sizes, int n_in,
                              void* d_out, int out_size, void* d_ws, size_t ws_size,
                              hipStream_t stream) {
  const float* x   = (const float*)d_in[0];   // [B, D]
  const float* c1  = (const float*)d_in[1];   // [H, D, 2]
  const float* c2  = (const float*)d_in[2];   // [D, H, 2]
  const int*   op1 = (const int*)d_in[3];     // [H]
  const int*   op2 = (const int*)d_in[4];     // [D]
  float* out = (float*)d_out;                 // [B, D]

  const int H = in_sizes[3];                  // 512
  const int D = in_sizes[4];                  // 1024
  const int B = in_sizes[0] / D;              // 256

  // Workspace layout: h [B,H] | thr1 [H,D] | thr2 [D,H]
  const size_t h_bytes   = (size_t)B * H * sizeof(float);
  const size_t thr_bytes = (size_t)H * D * sizeof(uint32_t);
  const size_t need      = h_bytes + 2 * thr_bytes;

  float* h = (float*)d_ws;
  dim3 blk(256);

  if (ws_size >= need) {
    uint32_t* thr1 = (uint32_t*)((char*)d_ws + h_bytes);
    uint32_t* thr2 = (uint32_t*)((char*)d_ws + h_bytes + thr_bytes);
    const int n_edges = H * D;
    ffc_thr_kernel<<<(n_edges + 255) / 256, blk, 0, stream>>>(
        (const float2*)c1, thr1, n_edges);
    ffc_thr_kernel<<<(n_edges + 255) / 256, blk, 0, stream>>>(
        (const float2*)c2, thr2, n_edges);
    ffc_layer_kernel<true><<<dim3(B, H / FFC_OPB), blk, 0, stream>>>(
        x, c1, thr1, op1, h, D, H, 0x9E3779B9u);
    ffc_layer_kernel<true><<<dim3(B, D / FFC_OPB), blk, 0, stream>>>(
        h, c2, thr2, op2, out, H, D, 0x85EBCA6Bu);
  } else {
    ffc_layer_kernel<false><<<dim3(B, H / FFC_OPB), blk, 0, stream>>>(
        x, c1, nullptr, op1, h, D, H, 0x9E3779B9u);
    ffc_layer_kernel<false><<<dim3(B, D / FFC_OPB), blk, 0, stream>>>(
        h, c2, nullptr, op2, out, H, D, 0x85EBCA6Bu);
  }
}